// SGDT_module_52329881534479
// MI455X (gfx1250) — compile-verified
//
#include <hip/hip_runtime.h>
#include <hip/hip_bf16.h>

#define N_TOK 4096
#define BB    16
#define CC    256
#define HH    128
#define DD    512   // 2*CC

// k-pair interleaved LDS panel strides (dwords per k-pair row).
// 2*ncols + 32 pad  ->  stride mod 64 == 32, so the two lane-halves
// (which differ by one k-pair row) hit disjoint bank sets.
#define W1P_STRIDE 288    // 2*128 + 32
#define WSP_STRIDE 1056   // 2*512 + 32

typedef __attribute__((ext_vector_type(2))) float v2f;
typedef __attribute__((ext_vector_type(8))) float v8f;

__device__ __forceinline__ v8f wmma_f32(v2f a, v2f b, v8f c) {
    // V_WMMA_F32_16X16X4_F32 : D(16x16 f32) = A(16x4 f32) * B(4x16 f32) + C
    return __builtin_amdgcn_wmma_f32_16x16x4_f32(
        /*neg_a=*/false, a, /*neg_b=*/false, b,
        /*c_mod=*/(short)0, c, /*reuse_a=*/false, /*reuse_b=*/false);
}

// ---------------------------------------------------------------------------
// K1: gate — sig[row] = sigmoid( relu(x[row]@w1 + b1) @ w2 + b2 ), row = n*B+b
// One wave handles a 16-row M-tile over all H=128 columns (8 WMMA col tiles).
// w1 staged in LDS with k-pair interleave: B frag = one ds_load_b64; all 8
// fragments batched ahead of the WMMA chain so DS latency is paid once.
// ---------------------------------------------------------------------------
__global__ __launch_bounds__(256) void gate_kernel(
    const float* __restrict__ x, const float* __restrict__ w1,
    const float* __restrict__ b1, const float* __restrict__ w2,
    const float* __restrict__ b2, float* __restrict__ sig)
{
    __shared__ float w1p[32 * W1P_STRIDE];   // 64-k panel, pair-interleaved (36 KB)
    const int tid  = threadIdx.x;
    const int lane = tid & 31, wave = tid >> 5;
    const int half = lane >> 4, l16 = lane & 15;
    const int row0 = (blockIdx.x * 8 + wave) * 16;   // 16 rows of (N*B, C)

    v8f acc[8];
#pragma unroll
    for (int t = 0; t < 8; ++t) acc[t] = (v8f)0.0f;

    const float* xrow = x + (size_t)(row0 + l16) * CC;

    for (int kp = 0; kp < CC; kp += 64) {
        __syncthreads();
        for (int e = tid; e < 64 * 128; e += 256) {
            int r = e >> 7, c = e & 127;
            w1p[(r >> 1) * W1P_STRIDE + 2 * c + (r & 1)] = w1[(size_t)(kp + r) * HH + c];
        }
        __syncthreads();
        for (int kk = 0; kk < 64; kk += 4) {
            v2f a = *(const v2f*)(xrow + kp + kk + 2 * half);
            const int kpair = (kk + 2 * half) >> 1;
            v2f bf[8];
#pragma unroll
            for (int t = 0; t < 8; ++t)
                bf[t] = *(const v2f*)&w1p[kpair * W1P_STRIDE + 2 * (t * 16 + l16)];
#pragma unroll
            for (int t = 0; t < 8; ++t)
                acc[t] = wmma_f32(a, bf[t], acc[t]);
        }
    }

    // logit[m] = sum_col relu(h[m,col]) * w2[col];  m = r + 8*half
    float part[8];
#pragma unroll
    for (int r = 0; r < 8; ++r) part[r] = 0.0f;
#pragma unroll
    for (int t = 0; t < 8; ++t) {
        int col = t * 16 + l16;
        float bb = b1[col], ww = w2[col];
#pragma unroll
        for (int r = 0; r < 8; ++r) {
            float h = acc[t][r] + bb;
            h = h > 0.0f ? h : 0.0f;
            part[r] += h * ww;
        }
    }
#pragma unroll
    for (int off = 1; off < 16; off <<= 1)
#pragma unroll
        for (int r = 0; r < 8; ++r) part[r] += __shfl_xor(part[r], off, 32);

    if (l16 == 0) {
        float bb2 = b2[0];
#pragma unroll
        for (int r = 0; r < 8; ++r) {
            float logit = part[r] + bb2;
            sig[row0 + r + 8 * half] = 1.0f / (1.0f + __expf(-logit));
        }
    }
}

// ---------------------------------------------------------------------------
// K2: per-batch counting + top-k selection (bitonic sort of (score,idx) keys)
// + index-ordered pairing lists slist/dlist and per-row flags.
// ---------------------------------------------------------------------------
__global__ __launch_bounds__(1024) void select_kernel(
    const float* __restrict__ sig, const int* __restrict__ mask,
    int* __restrict__ flags, int* __restrict__ slist, int* __restrict__ dlist,
    int* __restrict__ mincnt)
{
    const int b   = blockIdx.x;
    const int tid = threadIdx.x;
    __shared__ unsigned long long keys[N_TOK];   // 32 KB
    __shared__ unsigned char sel[N_TOK];         // 4 KB
    __shared__ int scanb[1024];                  // 4 KB
    __shared__ int cnts[2];
    __shared__ int kcnt_s;

    if (tid < 2) cnts[tid] = 0;
    __syncthreads();

    const int base = tid * 4;
    float sv[4]; bool dm[4], sm[4];
    int ld = 0, ls = 0;
#pragma unroll
    for (int q = 0; q < 4; ++q) {
        int n = base + q;
        float s = sig[n * BB + b];
        bool valid = (mask[b * N_TOK + n] == 0);
        bool fg = (s >= 0.5f);
        sv[q] = s; dm[q] = (!fg) && valid; sm[q] = fg && valid;
        ld += dm[q] ? 1 : 0;  ls += sm[q] ? 1 : 0;
        flags[n * BB + b] = 0;
    }
    atomicAdd(&cnts[0], ld);
    atomicAdd(&cnts[1], ls);
    __syncthreads();
    if (tid == 0) {
        int k = min(min(cnts[0], cnts[1]), 10000);
        kcnt_s = k; mincnt[b] = k;
    }
    __syncthreads();
    const int kcnt = kcnt_s;

    for (int g = 0; g < 2; ++g) {
        // build keys: ascending key == descending score, ties ascending index
#pragma unroll
        for (int q = 0; q < 4; ++q) {
            int n = base + q;
            bool member = (g == 0) ? dm[q] : sm[q];
            unsigned bits = __float_as_uint(sv[q]);   // sig in (0,1): order-preserving
            keys[n] = member
                ? ((((unsigned long long)(0xFFFFFFFFu - bits)) << 32) | (unsigned)n)
                : 0xFFFFFFFFFFFFFFFFull;
            sel[n] = 0;
        }
        __syncthreads();
        // bitonic sort ascending
        for (int k = 2; k <= N_TOK; k <<= 1) {
            for (int j = k >> 1; j > 0; j >>= 1) {
                for (int i = tid; i < N_TOK; i += 1024) {
                    int ixj = i ^ j;
                    if (ixj > i) {
                        unsigned long long a = keys[i], c = keys[ixj];
                        bool up = ((i & k) == 0);
                        if ((a > c) == up) { keys[i] = c; keys[ixj] = a; }
                    }
                }
                __syncthreads();
            }
        }
        // top-kcnt selected
        for (int i = tid; i < N_TOK; i += 1024)
            if (i < kcnt) sel[(unsigned)(keys[i] & 0xFFFFFFFFu)] = 1;
        __syncthreads();
        // block scan over sel -> ascending-index ranks
        int s0 = sel[base] + sel[base + 1] + sel[base + 2] + sel[base + 3];
        scanb[tid] = s0;
        __syncthreads();
        for (int off = 1; off < 1024; off <<= 1) {
            int add = (tid >= off) ? scanb[tid - off] : 0;
            __syncthreads();
            scanb[tid] += add;
            __syncthreads();
        }
        int r = scanb[tid] - s0;
#pragma unroll
        for (int q = 0; q < 4; ++q) {
            int n = base + q;
            if (sel[n]) {
                if (g == 0) { dlist[b * N_TOK + r] = n; flags[n * BB + b] |= 2; }
                else        { slist[b * N_TOK + r] = n; flags[n * BB + b] |= 1; }
                ++r;
            }
        }
        __syncthreads();
    }
}

// ---------------------------------------------------------------------------
// K3: out = (flags ? 0 : x)  — selected rows are fully overwritten by K4.
// ---------------------------------------------------------------------------
__global__ __launch_bounds__(256) void assemble_base(
    const float* __restrict__ x, const int* __restrict__ flags,
    float* __restrict__ out)
{
    size_t gid = (size_t)blockIdx.x * 256 + threadIdx.x;   // one float4 each
    int row = (int)(gid >> 6);                              // 64 float4 per row
    float4 v = ((const float4*)x)[gid];
    if (flags[row]) v = make_float4(0.f, 0.f, 0.f, 0.f);
    ((float4*)out)[gid] = v;
}

// ---------------------------------------------------------------------------
// K4: z = relu(x[split]@ws + bs); z1 -> split rows, z2 -> paired discard rows.
// Block = (tile of 16 split rows, batch). 8 waves x 64 output cols each.
// ws staged in LDS with k-pair interleave: B frag = one ds_load_b64; all 4
// fragments batched ahead of the WMMA chain.
// ---------------------------------------------------------------------------
__global__ __launch_bounds__(256) void pair_gemm(
    const float* __restrict__ x, const float* __restrict__ ws,
    const float* __restrict__ bs, const int* __restrict__ slist,
    const int* __restrict__ dlist, const int* __restrict__ mincnt,
    float* __restrict__ out)
{
    const int b    = blockIdx.y;
    const int tile = blockIdx.x;
    const int kcnt = mincnt[b];
    if (tile * 16 >= kcnt) return;          // uniform per block: EXEC stays full

    __shared__ float xs[16 * 260];           // gathered split rows of x (16.6 KB)
    __shared__ float wp[8 * WSP_STRIDE];     // ws k-panel, pair-interleaved (33.8 KB)
    __shared__ int srow[16], drow[16];

    const int tid = threadIdx.x, lane = tid & 31, wave = tid >> 5;
    const int half = lane >> 4, l16 = lane & 15;

    if (tid < 16) {
        int i  = tile * 16 + tid;
        int ic = (i < kcnt) ? i : (kcnt - 1);
        srow[tid] = slist[b * N_TOK + ic];
        drow[tid] = dlist[b * N_TOK + ic];
    }
    __syncthreads();

    for (int e = tid; e < 16 * 64; e += 256) {          // 16 rows x 64 float4
        int r = e >> 6, c4 = e & 63;
        float4 v = *(const float4*)(x + ((size_t)srow[r] * BB + b) * CC + c4 * 4);
        float* dst = &xs[r * 260 + c4 * 4];
        dst[0] = v.x; dst[1] = v.y; dst[2] = v.z; dst[3] = v.w;
    }

    v8f acc[4];
#pragma unroll
    for (int t = 0; t < 4; ++t) acc[t] = (v8f)0.0f;
    const int colbase0 = wave * 64;

    for (int kp = 0; kp < CC; kp += 16) {
        __syncthreads();
        for (int e = tid; e < 16 * 128; e += 256) {      // 16 rows x 128 float4
            int r = e >> 7, c4 = e & 127;
            float4 v = *(const float4*)(ws + (size_t)(kp + r) * DD + c4 * 4);
            float* dst = &wp[(r >> 1) * WSP_STRIDE + (r & 1)];
            dst[2 * (4 * c4 + 0)] = v.x;
            dst[2 * (4 * c4 + 1)] = v.y;
            dst[2 * (4 * c4 + 2)] = v.z;
            dst[2 * (4 * c4 + 3)] = v.w;
        }
        if (kp + 16 < CC)
            __builtin_prefetch(ws + (size_t)(kp + 16) * DD, 0, 1);
        __syncthreads();
        for (int kk = 0; kk < 16; kk += 4) {
            v2f a = *(const v2f*)&xs[l16 * 260 + kp + kk + 2 * half];
            const int kpair = (kk + 2 * half) >> 1;
            v2f bf[4];
#pragma unroll
            for (int t = 0; t < 4; ++t)
                bf[t] = *(const v2f*)&wp[kpair * WSP_STRIDE + 2 * (colbase0 + t * 16 + l16)];
#pragma unroll
            for (int t = 0; t < 4; ++t)
                acc[t] = wmma_f32(a, bf[t], acc[t]);
        }
    }

    const bool isz2 = (wave >= 4);
#pragma unroll
    for (int t = 0; t < 4; ++t) {
        int col = colbase0 + t * 16 + l16;               // 0..511
        float bias = bs[col];
#pragma unroll
        for (int r = 0; r < 8; ++r) {
            int m = r + 8 * half;
            int i = tile * 16 + m;
            if (i < kcnt) {
                float v = acc[t][r] + bias;
                v = v > 0.0f ? v : 0.0f;
                int orow = isz2 ? drow[m] : srow[m];
                int oc   = isz2 ? (col - CC) : col;
                out[((size_t)orow * BB + b) * CC + oc] = v;
            }
        }
    }
}

// ---------------------------------------------------------------------------
extern "C" void kernel_launch(void* const* d_in, const int* in_sizes, int n_in,
                              void* d_out, int out_size, void* d_ws, size_t ws_size,
                              hipStream_t stream) {
    const float* x    = (const float*)d_in[0];
    const int*   mask = (const int*)  d_in[1];
    const float* w1   = (const float*)d_in[2];
    const float* b1   = (const float*)d_in[3];
    const float* w2   = (const float*)d_in[4];
    const float* b2   = (const float*)d_in[5];
    const float* wsm  = (const float*)d_in[6];
    const float* bs   = (const float*)d_in[7];
    float* out = (float*)d_out;

    char* w = (char*)d_ws;
    float* sig    = (float*)(w);             // N*B floats      = 262144 B
    int*   flags  = (int*)  (w + 262144);    // N*B ints        = 262144 B
    int*   mincnt = (int*)  (w + 524288);    // 16 ints (padded)
    int*   dlist  = (int*)  (w + 528384);    // B*N ints        = 262144 B
    int*   slist  = (int*)  (w + 790528);    // B*N ints        = 262144 B

    gate_kernel  <<<dim3(512),      256,  0, stream>>>(x, w1, b1, w2, b2, sig);
    select_kernel<<<dim3(16),       1024, 0, stream>>>(sig, mask, flags, slist, dlist, mincnt);
    assemble_base<<<dim3(16384),    256,  0, stream>>>(x, flags, out);
    pair_gemm    <<<dim3(128, 16),  256,  0, stream>>>(x, wsm, bs, slist, dlist, mincnt, out);
}